// DictNet_2980707303969
// MI455X (gfx1250) — compile-verified
//
#include <hip/hip_runtime.h>
#include <math.h>

typedef float v2f __attribute__((ext_vector_type(2)));
typedef float v8f __attribute__((ext_vector_type(8)));

#define Nn 160            // nodes
#define Gg 64             // graphs
#define Ff 128            // features
#define Tt 21             // filters
#define LDS_LD 162        // padded LDS leading dim (bank-conflict avoidance, 8B aligned rows)
#define NS_ITERS 14       // Newton-Schulz iterations: 0.9969^(2^14) << fp32 eps

// ---------------------------------------------------------------------------
// One 16x16 output tile of Out = alpha*(A@B) + beta*I on V_WMMA_F32_16X16X4_F32.
// K=160 split into two independent accumulator chains (breaks the WMMA RAW
// dependency; only ~2 waves/SIMD at 300KB LDS/workgroup, so ILP matters).
// ---------------------------------------------------------------------------
__device__ __forceinline__ void wmma_tile(const float* __restrict__ A, int lda,
                                          const float* __restrict__ B, int ldb,
                                          float* __restrict__ Out, int ldo,
                                          float alpha, float beta,
                                          int mt, int nt, bool mirror,
                                          int l16, int half)
{
    v8f acc0 = {}, acc1 = {};

    // A fragment: lanes 0-15 -> rows mt..mt+15, K = k..k+1
    //             lanes 16-31 -> same rows,      K = k+2..k+3
    const float* arow = A + (mt + l16) * lda + (half << 1);
    // B fragment: lanes 0-15 -> cols nt..nt+15, K rows k..k+1
    //             lanes 16-31 -> same cols,      K rows k+2..k+3
    const float* bptr = B + (half << 1) * ldb + nt + l16;

    #pragma unroll 4
    for (int k = 0; k < 80; k += 4) {
        v2f af0, bf0, af1, bf1;
        af0.x = arow[k];
        af0.y = arow[k + 1];
        bf0.x = bptr[k * ldb];
        bf0.y = bptr[k * ldb + ldb];
        acc0 = __builtin_amdgcn_wmma_f32_16x16x4_f32(
            false, af0, false, bf0, (short)0, acc0, false, false);

        af1.x = arow[k + 80];
        af1.y = arow[k + 81];
        bf1.x = bptr[(k + 80) * ldb];
        bf1.y = bptr[(k + 80) * ldb + ldb];
        acc1 = __builtin_amdgcn_wmma_f32_16x16x4_f32(
            false, af1, false, bf1, (short)0, acc1, false, false);
    }

    // C/D layout: lane<16 -> rows mt..mt+7, lane>=16 -> rows mt+8..mt+15
    const int row0 = mt + (half << 3);
    const int col  = nt + l16;
    #pragma unroll
    for (int i = 0; i < 8; ++i) {
        const int r = row0 + i;
        const float v = alpha * (acc0[i] + acc1[i]) + ((r == col) ? beta : 0.0f);
        Out[r * ldo + col] = v;
        if (mirror) Out[col * ldo + r] = v;   // compile-time flag: no EXEC juggling
    }
}

// ---------------------------------------------------------------------------
// Symmetric-output 160x160 fp32 GEMM. All GEMM outputs in this pipeline are
// symmetric (products of commuting symmetric matrices: every Newton-Schulz
// iterate is a polynomial in B), so only the 55 upper-triangular 16x16 tiles
// are computed (1.8x WMMA reduction). Diagonal and off-diagonal tiles are
// handled in separate wave-uniform loops so the mirror store is unconditional
// (no per-lane saveexec in the epilogue).
// 256 threads = 8 waves; EXEC all ones at every WMMA.
// ---------------------------------------------------------------------------
__device__ __forceinline__ void gemm160_sym(const float* __restrict__ A, int lda,
                                            const float* __restrict__ B, int ldb,
                                            float* __restrict__ Out, int ldo,
                                            float alpha, float beta,
                                            int wave, int lane)
{
    const int l16  = lane & 15;
    const int half = lane >> 4;

    // 10 diagonal tiles (ti == tj): no mirror
    for (int d = wave; d < 10; d += 8) {
        wmma_tile(A, lda, B, ldb, Out, ldo, alpha, beta,
                  d << 4, d << 4, false, l16, half);
    }
    // 45 strictly-upper tiles (ti < tj): unconditional mirror
    for (int o = wave; o < 45; o += 8) {
        int ti = 0, rem = o;
        while (rem >= 9 - ti) { rem -= 9 - ti; ++ti; }
        const int tj = ti + 1 + rem;
        wmma_tile(A, lda, B, ldb, Out, ldo, alpha, beta,
                  ti << 4, tj << 4, true, l16, half);
    }
}

// ---------------------------------------------------------------------------
// Kernel 1: normalized Laplacian per graph. L = I - D^-1/2 A D^-1/2
// ---------------------------------------------------------------------------
__global__ void k1_laplacian(const float* __restrict__ A,
                             float* __restrict__ wsL)
{
    const int g   = blockIdx.x;
    const int tid = threadIdx.x;
    __shared__ float dis[Nn];

    const float* Ag = A + (size_t)g * Nn * Nn;
    for (int n = tid; n < Nn; n += blockDim.x) {
        float s = 0.f;
        for (int m = 0; m < Nn; ++m) s += Ag[n * Nn + m];
        dis[n] = (s > 0.f) ? rsqrtf(s) : 0.f;
    }
    __syncthreads();

    float* Lg = wsL + (size_t)g * Nn * Nn;
    for (int idx = tid; idx < Nn * Nn; idx += blockDim.x) {
        const int r = idx / Nn, c = idx % Nn;
        Lg[idx] = ((r == c) ? 1.f : 0.f) - dis[r] * Ag[idx] * dis[c];
    }
}

// ---------------------------------------------------------------------------
// Kernel 2: per (g,t) spectral chain, fully LDS-resident WMMA GEMMs.
//   M = L - t*0.1*I ; M2 = M@M ; B = 40*(M2@M2) + I
//   Binv via Newton-Schulz  X <- X(2I - BX),  X0 = (2/642) I
//     (eig(L) in [0,2] => eig(B) in [1,641] => ||I - c0 B|| <= 0.9969,
//      quadratic convergence: residual^(2^14) is below fp32 noise)
//   emits weighted column-sums of Fmat = Binv@Binv via  (Binv 1)^T Binv
// ---------------------------------------------------------------------------
__global__ void k2_spectral(const float* __restrict__ wsL,
                            const float* __restrict__ C,
                            float* __restrict__ wsB,
                            float* __restrict__ wsCS)
{
    extern __shared__ float smem[];
    float* buf0 = smem;
    float* buf1 = smem + Nn * LDS_LD;
    float* buf2 = smem + 2 * Nn * LDS_LD;

    const int b    = blockIdx.x;
    const int g    = b / Tt;
    const int t    = b % Tt;
    const int tid  = threadIdx.x;
    const int lane = tid & 31;
    const int wave = tid >> 5;

    // normalized filter weight Cn[t]
    float nrm2 = 0.f;
    #pragma unroll
    for (int i = 0; i < Tt; ++i) { const float c = C[i]; nrm2 += c * c; }
    const float wt = C[t] / fmaxf(sqrtf(nrm2), 1e-12f);

    // buf0 = M = L[g] - t*STEP*I
    const float* Lg = wsL + (size_t)g * Nn * Nn;
    const float tshift = (float)t * 0.1f;
    for (int idx = tid; idx < Nn * Nn; idx += blockDim.x) {
        const int r = idx / Nn, c = idx % Nn;
        float v = Lg[idx];
        if (r == c) v -= tshift;
        buf0[r * LDS_LD + c] = v;
    }
    __syncthreads();

    gemm160_sym(buf0, LDS_LD, buf0, LDS_LD, buf1, LDS_LD, 1.f, 0.f, wave, lane);  // M2
    __syncthreads();
    gemm160_sym(buf1, LDS_LD, buf1, LDS_LD, buf2, LDS_LD, 40.f, 1.f, wave, lane); // B
    __syncthreads();

    // spill B to its L2-resident slab so all 3 LDS buffers are free to ping-pong
    float* gB = wsB + (size_t)b * Nn * Nn;
    for (int idx = tid; idx < Nn * Nn; idx += blockDim.x)
        gB[idx] = buf2[(idx / Nn) * LDS_LD + (idx % Nn)];
    __threadfence();
    __syncthreads();

    // warm the near caches for the iterated A-operand reads (global_prefetch_b8)
    for (int idx = tid * 64; idx < Nn * Nn; idx += blockDim.x * 64)
        __builtin_prefetch(gB + idx, 0, 3);

    // X0 = (2 / (lam_min + lam_max)) I  with eig(B) in [1, 641]
    const float c0 = 2.0f / 642.0f;
    for (int idx = tid; idx < Nn * Nn; idx += blockDim.x) {
        const int r = idx / Nn, c = idx % Nn;
        buf0[r * LDS_LD + c] = (r == c) ? c0 : 0.f;
    }
    __syncthreads();

    float* X  = buf0;
    float* Xn = buf2;
    float* Tm = buf1;
    for (int it = 0; it < NS_ITERS; ++it) {
        // Tm = 2I - B @ X    (A-operand streamed from L2-resident slab)
        gemm160_sym(gB, Nn, X, LDS_LD, Tm, LDS_LD, -1.f, 2.f, wave, lane);
        __syncthreads();
        // Xn = X @ Tm
        gemm160_sym(X, LDS_LD, Tm, LDS_LD, Xn, LDS_LD, 1.f, 0.f, wave, lane);
        __syncthreads();
        float* tmp = X; X = Xn; Xn = tmp;
    }
    // X == Binv (buf0 after an even number of swaps; Tm==buf1 never aliases X)

    // u = Binv @ 1  (== column sums, Binv symmetric)
    float* u = Tm;
    for (int k = tid; k < Nn; k += blockDim.x) {
        float s = 0.f;
        for (int n = 0; n < Nn; ++n) s += X[n * LDS_LD + k];
        u[k] = s;
    }
    __syncthreads();

    // colsum(Fmat)[m] = sum_k u[k] * Binv[k][m];  store weighted slice
    float* cs = wsCS + (size_t)b * Nn;
    for (int m = tid; m < Nn; m += blockDim.x) {
        float s = 0.f;
        for (int k = 0; k < Nn; ++k) s += u[k] * X[k * LDS_LD + m];
        cs[m] = wt * s;
    }
}

// ---------------------------------------------------------------------------
// Kernel 3: emb[g,f] = (1/N) * sum_m (1 - colsumL[g,m]) * x[g,m,f]
// (uses mean_rows((I - L_hat) @ x) == (colsum(A_hat)/N) @ x)
// ---------------------------------------------------------------------------
__global__ void k3_embed(const float* __restrict__ wsCS,
                         const float* __restrict__ x,
                         float* __restrict__ emb)
{
    const int g   = blockIdx.x;
    const int tid = threadIdx.x;        // 160 threads
    __shared__ float w[Nn];

    if (tid < Nn) {
        float s = 0.f;
        for (int t = 0; t < Tt; ++t)
            s += wsCS[((size_t)g * Tt + t) * Nn + tid];
        w[tid] = 1.f - s;               // column sum of A_hat
    }
    __syncthreads();

    if (tid < Ff) {
        float acc = 0.f;
        const float* xg = x + (size_t)g * Nn * Ff + tid;
        for (int m = 0; m < Nn; ++m) acc += w[m] * xg[m * Ff];
        emb[g * Ff + tid] = acc * (1.0f / (float)Nn);
    }
}

// ---------------------------------------------------------------------------
// Kernel 4: pairwise distances + class-mask contrast terms + sparsity -> scalar
// ---------------------------------------------------------------------------
__global__ void k4_final(const float* __restrict__ emb,
                         const float* __restrict__ C,
                         const int* __restrict__ y,
                         float* __restrict__ out)
{
    __shared__ float s00[Gg], s11[Gg], s01[Gg];
    const int i = threadIdx.x;          // 64 threads
    const float* ei = emb + i * Ff;
    const int yi = y[i];

    float a00 = 0.f, a11 = 0.f, a01 = 0.f;
    for (int j = 0; j < Gg; ++j) {
        const float* ej = emb + j * Ff;
        float sq = 0.f;
        for (int f = 0; f < Ff; ++f) { const float d = ei[f] - ej[f]; sq += d * d; }
        const float dist = (sq > 0.f) ? sqrtf(sq) : 0.f;
        const int yj = y[j];
        if (yi == 0 && yj == 0) a00 += dist;
        if (yi == 1 && yj == 1) a11 += dist;
        if (yi == 0 && yj == 1) a01 += dist;
    }
    s00[i] = a00; s11[i] = a11; s01[i] = a01;
    __syncthreads();

    if (i == 0) {
        float S00 = 0.f, S11 = 0.f, S01 = 0.f;
        int n0 = 0, n1 = 0;
        for (int k = 0; k < Gg; ++k) {
            S00 += s00[k]; S11 += s11[k]; S01 += s01[k];
            if (y[k] == 0) n0++; else if (y[k] == 1) n1++;
        }
        const float fn0 = (float)n0, fn1 = (float)n1;
        const float pos  = S00 / (fn0 * fn0) + S11 / (fn1 * fn1);
        const float beta = fn0 * fn1 * 0.5f + 1e-13f;
        const float neg  = -0.5f * S01 / beta;

        float sabs = 0.f, sq = 0.f;
        for (int k = 0; k < Tt; ++k) { const float c = C[k]; sabs += fabsf(c); sq += c * c; }
        const float nrm      = sqrtf(sq);
        const float dims     = sqrtf(21.0f);
        const float sparsity = (dims - sabs / nrm) / (dims - 1.0f);

        out[0] = sparsity + pos + neg;
    }
}

// ---------------------------------------------------------------------------
extern "C" void kernel_launch(void* const* d_in, const int* in_sizes, int n_in,
                              void* d_out, int out_size, void* d_ws, size_t ws_size,
                              hipStream_t stream)
{
    (void)in_sizes; (void)n_in; (void)out_size; (void)ws_size;

    const float* A = (const float*)d_in[0];   // (64,160,160)
    const float* x = (const float*)d_in[1];   // (64,160,128)
    const float* C = (const float*)d_in[2];   // (21,1)
    const int*   y = (const int*)d_in[3];     // (64,1)
    float* out = (float*)d_out;               // scalar

    float* ws   = (float*)d_ws;
    float* wsL  = ws;                                   // 64*160*160   (6.6 MB)
    float* wsB  = wsL + (size_t)Gg * Nn * Nn;           // 1344*160*160 (137.6 MB, L2-resident)
    float* wsCS = wsB + (size_t)Gg * Tt * Nn * Nn;      // 1344*160
    float* emb  = wsCS + (size_t)Gg * Tt * Nn;          // 64*128

    k1_laplacian<<<Gg, 256, 0, stream>>>(A, wsL);

    const size_t lds_bytes = (size_t)3 * Nn * LDS_LD * sizeof(float); // 311040 B <= 320 KB WGP LDS
    k2_spectral<<<Gg * Tt, 256, lds_bytes, stream>>>(wsL, C, wsB, wsCS);

    k3_embed<<<Gg, Nn, 0, stream>>>(wsCS, x, emb);
    k4_final<<<1, Gg, 0, stream>>>(emb, C, y, out);
}